// CenterNet_49984829390856
// MI455X (gfx1250) — compile-verified
//
#include <hip/hip_runtime.h>
#include <hip/hip_bf16.h>
#include <stdint.h>

// CenterNet decode for MI455X (gfx1250, wave32).
// Bandwidth-bound: ~180 MB traffic -> ~7.7 us floor @ 23.3 TB/s.
// No GEMM content -> no WMMA; CDNA5 paths used: NT loads, global_prefetch,
// global_load_async_to_lds_b128 + s_wait_asynccnt for LDS staging of top-k.

#define B   32
#define C   80
#define H   128
#define W   128
#define HW  (H * W)
#define KTOP 100
#define WH_SCALE 16.0f
#define STRIDE_F 4.0f

typedef float v4f __attribute__((ext_vector_type(4)));
typedef int   v4i __attribute__((ext_vector_type(4)));

// ---------------------------------------------------------------------------
// Kernel 1: per-pixel max + argmax over 80 channels, sigmoid.
// One thread = 4 consecutive pixels (float4 loads, fully coalesced within a
// channel plane). Heatmap is streamed once -> non-temporal loads so the
// 168 MB stream does not wipe L2. Channel stride is 64 KB, so issue explicit
// global_prefetch a few channels ahead.
// ---------------------------------------------------------------------------
__global__ __launch_bounds__(256)
void k_chanmax(const float* __restrict__ hm,
               float* __restrict__ scores,
               int*   __restrict__ cls) {
    const int t     = blockIdx.x * blockDim.x + threadIdx.x;   // 4-pixel groups
    const int total = B * (HW / 4);
    if (t >= total) return;
    const int b  = t / (HW / 4);
    const int p4 = t - b * (HW / 4);

    const v4f* base = (const v4f*)(hm + (size_t)b * C * HW) + p4;

    v4f m = __builtin_nontemporal_load(base);   // channel 0
    v4i am; am.x = 0; am.y = 0; am.z = 0; am.w = 0;

#pragma unroll 4
    for (int c = 1; c < C; ++c) {
        if (c + 8 < C)
            __builtin_prefetch((const void*)(base + (size_t)(c + 8) * (HW / 4)), 0, 1);
        v4f v = __builtin_nontemporal_load(base + (size_t)c * (HW / 4));
        if (v.x > m.x) { m.x = v.x; am.x = c; }
        if (v.y > m.y) { m.y = v.y; am.y = c; }
        if (v.z > m.z) { m.z = v.z; am.z = c; }
        if (v.w > m.w) { m.w = v.w; am.w = c; }
    }

    v4f s;
    s.x = 1.0f / (1.0f + __expf(-m.x));
    s.y = 1.0f / (1.0f + __expf(-m.y));
    s.z = 1.0f / (1.0f + __expf(-m.z));
    s.w = 1.0f / (1.0f + __expf(-m.w));

    ((v4f*)(scores + (size_t)b * HW))[p4] = s;
    ((v4i*)(cls    + (size_t)b * HW))[p4] = am;
}

// ---------------------------------------------------------------------------
// Kernel 2: 3x3 NMS: keep score where it equals the 3x3 max (SAME padding
// with -inf == just skip out-of-bounds taps), else 0.
// ---------------------------------------------------------------------------
__global__ __launch_bounds__(256)
void k_nms(const float* __restrict__ scores, float* __restrict__ nms) {
    const int t = blockIdx.x * blockDim.x + threadIdx.x;
    if (t >= B * HW) return;
    const int b = t >> 14;          // / HW
    const int p = t & (HW - 1);
    const int y = p >> 7;           // / W
    const int x = p & (W - 1);

    const float* sb = scores + (size_t)b * HW;
    const float  s  = sb[p];
    float mx = s;
#pragma unroll
    for (int dy = -1; dy <= 1; ++dy) {
        const int yy = y + dy;
        if (yy < 0 || yy >= H) continue;
#pragma unroll
        for (int dx = -1; dx <= 1; ++dx) {
            const int xx = x + dx;
            if (xx < 0 || xx >= W) continue;
            const float v = sb[yy * W + xx];
            mx = fmaxf(mx, v);
        }
    }
    nms[t] = (s == mx) ? s : 0.0f;
}

// ---------------------------------------------------------------------------
// Kernel 3: per-batch top-100. One 1024-thread block (32 waves) per batch.
// Stage the 64 KB score map into LDS with CDNA5 async global->LDS b128 copies
// (ASYNCcnt path), then run 100 iterative max-reductions over LDS using a
// packed u64 key: (value_bits << 32) | (HW-1-index). All surviving values are
// >= 0 so their IEEE bits order monotonically; the reversed index gives
// lax.top_k's lowest-index-wins tie break. Selected entries are overwritten
// with -1.0f whose key maps to 0 (never re-selected).
// Batch 0 additionally records its index list (the reference reuses
// top_index[0] for every batch's boxes/labels).
// ---------------------------------------------------------------------------
__global__ __launch_bounds__(1024)
void k_topk(const float* __restrict__ nms,
            float* __restrict__ topscore,   // [B*KTOP]
            int*   __restrict__ topidx0) {  // [KTOP]
    __shared__ __align__(16) float sv[HW];                 // 64 KB
    __shared__ unsigned long long sred[1024];              // 8 KB

    const int b   = blockIdx.x;
    const int tid = threadIdx.x;
    const float* src = nms + (size_t)b * HW;

    // Async copy global -> LDS, 16 B per op per lane.
    for (int i = tid * 4; i < HW; i += 1024 * 4) {
        unsigned lds_off = (unsigned)(size_t)&sv[i];
        unsigned long long ga = (unsigned long long)(uintptr_t)(src + i);
        asm volatile("global_load_async_to_lds_b128 %0, %1, off"
                     :: "v"(lds_off), "v"(ga) : "memory");
    }
    asm volatile("s_wait_asynccnt 0" ::: "memory");
    __syncthreads();

    for (int k = 0; k < KTOP; ++k) {
        unsigned long long best = 0ull;
        for (int i = tid; i < HW; i += 1024) {
            const unsigned bits = __float_as_uint(sv[i]);
            const unsigned long long key =
                (bits & 0x80000000u)
                    ? 0ull
                    : (((unsigned long long)bits << 32) |
                       (unsigned)(HW - 1 - i));
            if (key > best) best = key;
        }
        sred[tid] = best;
        __syncthreads();
#pragma unroll
        for (int s = 512; s > 0; s >>= 1) {
            if (tid < s) {
                if (sred[tid + s] > sred[tid]) sred[tid] = sred[tid + s];
            }
            __syncthreads();
        }
        const unsigned long long w = sred[0];
        const float v  = __uint_as_float((unsigned)(w >> 32));
        const int  idx = (HW - 1) - (int)(w & 0xFFFFFFFFu);
        if (tid == 0) {
            topscore[b * KTOP + k] = v;
            if (b == 0) topidx0[k] = idx;
            sv[idx] = -1.0f;       // key -> 0, removed from contention
        }
        __syncthreads();
    }
}

// ---------------------------------------------------------------------------
// Kernel 4: emit (B, 100, 6). Boxes/labels gathered through batch-0 indices
// (reference quirk), scores are per-batch.
// box = [x - 16*wh0, y - 16*wh1, x + 16*wh2, y + 16*wh3]
// ---------------------------------------------------------------------------
__global__ __launch_bounds__(128)
void k_emit(const float* __restrict__ wh,
            const int*   __restrict__ cls,
            const float* __restrict__ topscore,
            const int*   __restrict__ topidx0,
            float* __restrict__ out) {
    const int t = blockIdx.x * blockDim.x + threadIdx.x;
    if (t >= B * KTOP) return;
    const int b = t / KTOP;
    const int k = t - b * KTOP;

    const int idx = topidx0[k];
    const float fx = (float)(idx & (W - 1)) * STRIDE_F;
    const float fy = (float)(idx >> 7)      * STRIDE_F;

    const float* whb = wh + (size_t)b * 4 * HW + idx;
    const float w0 = whb[0 * HW] * WH_SCALE;
    const float w1 = whb[1 * HW] * WH_SCALE;
    const float w2 = whb[2 * HW] * WH_SCALE;
    const float w3 = whb[3 * HW] * WH_SCALE;

    float* o = out + (size_t)t * 6;
    o[0] = fx - w0;
    o[1] = fy - w1;
    o[2] = fx + w2;
    o[3] = fy + w3;
    o[4] = topscore[t];
    o[5] = (float)cls[(size_t)b * HW + idx];
}

// ---------------------------------------------------------------------------
extern "C" void kernel_launch(void* const* d_in, const int* in_sizes, int n_in,
                              void* d_out, int out_size, void* d_ws, size_t ws_size,
                              hipStream_t stream) {
    const float* hm = (const float*)d_in[0];   // (32,80,128,128)
    const float* wh = (const float*)d_in[1];   // (32,4,128,128)
    float* out = (float*)d_out;                // (32,100,6)

    const size_t SZ = (size_t)B * HW * sizeof(float);   // 2 MB
    char* ws = (char*)d_ws;
    float* scores   = (float*)(ws);                     // B*HW f32
    int*   cls      = (int*)  (ws + SZ);                // B*HW i32
    float* nms      = (float*)(ws + 2 * SZ);            // B*HW f32
    float* topscore = (float*)(ws + 3 * SZ);            // B*KTOP f32
    int*   topidx0  = (int*)  (ws + 3 * SZ + 16384);    // KTOP i32

    {   // kernel 1: channel max/argmax + sigmoid
        const int total  = B * (HW / 4);
        const int blocks = (total + 255) / 256;
        k_chanmax<<<blocks, 256, 0, stream>>>(hm, scores, cls);
    }
    {   // kernel 2: 3x3 NMS
        const int total  = B * HW;
        const int blocks = (total + 255) / 256;
        k_nms<<<blocks, 256, 0, stream>>>(scores, nms);
    }
    {   // kernel 3: per-batch top-100 (async LDS staging)
        k_topk<<<B, 1024, 0, stream>>>(nms, topscore, topidx0);
    }
    {   // kernel 4: gather + emit
        const int total  = B * KTOP;
        const int blocks = (total + 127) / 128;
        k_emit<<<blocks, 128, 0, stream>>>(wh, cls, topscore, topidx0, out);
    }
}